// DeformableSlidingWindowAttentionGuided3D_46359876993359
// MI455X (gfx1250) — compile-verified
//
#include <hip/hip_runtime.h>
#include <math.h>

typedef __attribute__((ext_vector_type(2))) float v2f;
typedef __attribute__((ext_vector_type(8))) float v8f;

#define B_   2
#define C_   192
#define NH_  6
#define HD_  32
#define DD_  16
#define HH_  32
#define WW_  32
#define S_   16384            // DD_*HH_*WW_
#define SCALE_ 0.17677669529663687f   // 32^-0.5
#define LNEPS_ 1e-5f

#define PLANE_ ((size_t)6291456)      // B_*C_*S_ floats (also B_*NH_*S_*HD_)

// ======================= fused q/k/v 1x1-conv GEMM ==========================
// Out[b,m,s] = sum_c W[m,c] * X[b,c,s] + bias[m]
// One wave computes one 16(M)x16(N) tile for all three weight sets,
// chaining 48 V_WMMA_F32_16X16X4_F32 ops per accumulator.
// q written channel-major [b,c,s] (needed by dwconv); k/v written channel-last
// [b,head,s,32] so attention trilinear taps are contiguous 128B lines.
__global__ __launch_bounds__(256) void qkv_gemm_kernel(
    const float* __restrict__ x,
    const float* __restrict__ qw, const float* __restrict__ qbias,
    const float* __restrict__ kw, const float* __restrict__ kbias,
    const float* __restrict__ vw, const float* __restrict__ vbias,
    float* __restrict__ q_std, float* __restrict__ k_t, float* __restrict__ v_t)
{
  const int lane = threadIdx.x & 31;
  const int gw   = blockIdx.x * 8 + (threadIdx.x >> 5);
  if (gw >= B_ * 12 * 1024) return;
  const int nt = gw & 1023;              // 1024 N tiles of 16 spatial
  const int mt = (gw >> 10) % 12;        // 12 M tiles of 16 channels
  const int b  = gw / (12 * 1024);
  const int n     = lane & 15;           // N col (B/C/D) == A row M
  const int khalf = (lane >> 4) << 1;    // K group: lanes 0-15 -> {0,1}, 16-31 -> {2,3}
  const int sbase = nt << 4;
  const int mbase = mt << 4;

  const float* xb  = x  + (size_t)b * C_ * S_ + sbase + n;
  const float* qwr = qw + (size_t)(mbase + n) * C_ + khalf;
  const float* kwr = kw + (size_t)(mbase + n) * C_ + khalf;
  const float* vwr = vw + (size_t)(mbase + n) * C_ + khalf;

  v8f accq = {}; v8f acck = {}; v8f accv = {};
  for (int k0 = 0; k0 < C_; k0 += 4) {
    if (k0 + 16 < C_) __builtin_prefetch(xb + (size_t)(k0 + 16 + khalf) * S_, 0, 1);
    v2f bx;
    bx.x = xb[(size_t)(k0 + khalf) * S_];
    bx.y = xb[(size_t)(k0 + khalf + 1) * S_];
    v2f aq = *(const v2f*)(qwr + k0);
    v2f ak = *(const v2f*)(kwr + k0);
    v2f av = *(const v2f*)(vwr + k0);
    accq = __builtin_amdgcn_wmma_f32_16x16x4_f32(false, aq, false, bx, (short)0, accq, false, false);
    acck = __builtin_amdgcn_wmma_f32_16x16x4_f32(false, ak, false, bx, (short)0, acck, false, false);
    accv = __builtin_amdgcn_wmma_f32_16x16x4_f32(false, av, false, bx, (short)0, accv, false, false);
  }
  const int mhi = (lane >> 4) << 3;      // C/D: M = r + 8*(lane/16)
  const int ss  = sbase + n;
  for (int r = 0; r < 8; ++r) {
    int mm = mbase + r + mhi;
    q_std[((size_t)b * C_ + mm) * S_ + ss] = accq[r] + qbias[mm];
    size_t tix = (((size_t)b * NH_ + (mm >> 5)) * S_ + ss) * HD_ + (mm & 31);
    k_t[tix] = acck[r] + kbias[mm];
    v_t[tix] = accv[r] + vbias[mm];
  }
}

// ================= fused depthwise 3x3x3 convs (t_raw, lepe) ================
__global__ __launch_bounds__(256) void dwconv_kernel(
    const float* __restrict__ q_std,
    const float* __restrict__ odw,  const float* __restrict__ odb,
    const float* __restrict__ rpew, const float* __restrict__ rpeb,
    float* __restrict__ t, float* __restrict__ lepe)
{
  int idx = blockIdx.x * 256 + threadIdx.x;
  if (idx >= B_ * C_ * S_) return;
  int xw = idx & 31, yh = (idx >> 5) & 31, zd = (idx >> 10) & 15;
  int c  = (idx >> 14) % C_;
  int b  = idx / (C_ * S_);
  const float* qp = q_std + ((size_t)b * C_ + c) * S_;
  const float* w1 = odw  + c * 27;
  const float* w2 = rpew + c * 27;
  float s1 = 0.f, s2 = 0.f;
  for (int kz = 0; kz < 3; ++kz) {
    int z = zd + kz - 1; if (z < 0 || z >= DD_) continue;
    for (int ky = 0; ky < 3; ++ky) {
      int y = yh + ky - 1; if (y < 0 || y >= HH_) continue;
      for (int kx = 0; kx < 3; ++kx) {
        int xx = xw + kx - 1; if (xx < 0 || xx >= WW_) continue;
        float v = qp[(z * HH_ + y) * WW_ + xx];
        int wi = (kz * 3 + ky) * 3 + kx;
        s1 += w1[wi] * v; s2 += w2[wi] * v;
      }
    }
  }
  t[idx]    = s1 + odb[c];
  lepe[idx] = s2 + rpeb[c];
}

// ================== channel LayerNorm + exact GELU (in place) ===============
// One wave per (b, s); 192 channels = 6 per lane; shfl_xor butterflies.
__global__ __launch_bounds__(256) void ln_gelu_kernel(
    float* __restrict__ t, const float* __restrict__ lnw, const float* __restrict__ lnb)
{
  int lane = threadIdx.x & 31;
  int p = blockIdx.x * 8 + (threadIdx.x >> 5);
  if (p >= B_ * S_) return;
  int b = p / S_, s = p % S_;
  float* tp = t + (size_t)b * C_ * S_ + s;
  float vals[6];
  float sum = 0.f;
  for (int i = 0; i < 6; ++i) { vals[i] = tp[(size_t)(lane + 32 * i) * S_]; sum += vals[i]; }
  for (int o = 16; o; o >>= 1) sum += __shfl_xor(sum, o, 32);
  float mean = sum * (1.f / 192.f);
  float vs = 0.f;
  for (int i = 0; i < 6; ++i) { float d = vals[i] - mean; vs += d * d; }
  for (int o = 16; o; o >>= 1) vs += __shfl_xor(vs, o, 32);
  float inv = rsqrtf(vs * (1.f / 192.f) + LNEPS_);
  for (int i = 0; i < 6; ++i) {
    int c = lane + 32 * i;
    float xn = (vals[i] - mean) * inv * lnw[c] + lnb[c];
    tp[(size_t)c * S_] = 0.5f * xn * (1.f + erff(xn * 0.70710678118654752f));
  }
}

// ============ offset projection (18x192 GEMM) + tanh + scale ================
// offs[b,head,s,axis] = tanh(sum_c opw[head*3+axis,c]*t[b,c,s] + opb) * 2/(dim-1)
__global__ __launch_bounds__(256) void offs_kernel(
    const float* __restrict__ t, const float* __restrict__ opw,
    const float* __restrict__ opb, float* __restrict__ offs_o)
{
  int idx = blockIdx.x * 256 + threadIdx.x;
  if (idx >= 18 * B_ * S_) return;
  int s = idx % S_;
  int b = (idx / S_) % B_;
  int o = idx / (S_ * B_);
  const float* tp = t + (size_t)b * C_ * S_ + s;
  const float* wr = opw + o * C_;
  float acc = opb[o];
  for (int c = 0; c < C_; ++c) acc += wr[c] * tp[(size_t)c * S_];
  int head = o / 3, axis = o % 3;
  float sc = (axis == 0) ? (2.f / (WW_ - 1)) : (axis == 1 ? (2.f / (HH_ - 1)) : (2.f / (DD_ - 1)));
  offs_o[(((size_t)b * NH_ + head) * S_ + s) * 3 + axis] = tanhf(acc) * sc;
}

// ==================== deformable attention (online softmax) =================
__device__ __forceinline__ float trisample(const float* __restrict__ base,
                                           int z0, int z1, int y0, int y1, int x0, int x1,
                                           float wz, float wy, float wx)
{
  float v000 = base[(size_t)((z0 * HH_ + y0) * WW_ + x0) * HD_];
  float v001 = base[(size_t)((z0 * HH_ + y0) * WW_ + x1) * HD_];
  float v010 = base[(size_t)((z0 * HH_ + y1) * WW_ + x0) * HD_];
  float v011 = base[(size_t)((z0 * HH_ + y1) * WW_ + x1) * HD_];
  float v100 = base[(size_t)((z1 * HH_ + y0) * WW_ + x0) * HD_];
  float v101 = base[(size_t)((z1 * HH_ + y0) * WW_ + x1) * HD_];
  float v110 = base[(size_t)((z1 * HH_ + y1) * WW_ + x0) * HD_];
  float v111 = base[(size_t)((z1 * HH_ + y1) * WW_ + x1) * HD_];
  float c00 = v000 + (v001 - v000) * wx;
  float c01 = v010 + (v011 - v010) * wx;
  float c10 = v100 + (v101 - v100) * wx;
  float c11 = v110 + (v111 - v110) * wx;
  float c0 = c00 + (c01 - c00) * wy;
  float c1 = c10 + (c11 - c10) * wy;
  return c0 + (c1 - c0) * wz;
}

__global__ __launch_bounds__(256) void attn_kernel(
    const float* __restrict__ q_std, const float* __restrict__ k_t,
    const float* __restrict__ v_t,  const float* __restrict__ offs_o,
    const float* __restrict__ bias0, const float* __restrict__ bias1,
    float* __restrict__ attn_t)
{
  int lane = threadIdx.x & 31;                 // lane == head-dim channel
  int wv = blockIdx.x * 8 + (threadIdx.x >> 5);
  if (wv >= B_ * NH_ * S_) return;
  int s    = wv % S_;
  int head = (wv / S_) % NH_;
  int b    = wv / (S_ * NH_);
  int branch = head / 3, hl = head % 3;
  float dil = branch ? 2.f : 1.f;
  const float* bias = (branch ? bias1 : bias0) + hl * 27;

  int xw = s & 31, yh = (s >> 5) & 31, zd = s >> 10;
  float qv = q_std[((size_t)b * C_ + head * HD_ + lane) * S_ + s] * SCALE_;
  const float* op = offs_o + (((size_t)b * NH_ + head) * S_ + s) * 3;
  float ox = op[0], oy = op[1], oz = op[2];
  const float nx = 2.f / (WW_ - 1), ny = 2.f / (HH_ - 1), nz = 2.f / (DD_ - 1);
  float gx0 = (-1.f + 2.f * xw / (WW_ - 1)) + ox;
  float gy0 = (-1.f + 2.f * yh / (HH_ - 1)) + oy;
  float gz0 = (-1.f + 2.f * zd / (DD_ - 1)) + oz;
  size_t hbase = (((size_t)b * NH_ + head) * S_) * HD_ + lane;
  const float* kb = k_t + hbase;
  const float* vb = v_t + hbase;

  float m = -1e30f, l = 0.f, acc = 0.f;       // online softmax state
  for (int o = 0; o < 27; ++o) {
    float rx = (float)(o % 3 - 1) * dil;      // rel tuple order (x,y,z), idx = (z*3+y)*3+x
    float ry = (float)((o / 3) % 3 - 1) * dil;
    float rz = (float)(o / 9 - 1) * dil;
    float gx = fminf(fmaxf(gx0 + rx * nx, -1.f), 1.f);
    float gy = fminf(fmaxf(gy0 + ry * ny, -1.f), 1.f);
    float gz = fminf(fmaxf(gz0 + rz * nz, -1.f), 1.f);
    float fx = (gx + 1.f) * 0.5f * (WW_ - 1);
    float fy = (gy + 1.f) * 0.5f * (HH_ - 1);
    float fz = (gz + 1.f) * 0.5f * (DD_ - 1);
    int x0 = (int)floorf(fx); float wx = fx - x0; int x1 = min(x0 + 1, WW_ - 1);
    int y0 = (int)floorf(fy); float wy = fy - y0; int y1 = min(y0 + 1, HH_ - 1);
    int z0 = (int)floorf(fz); float wz = fz - z0; int z1 = min(z0 + 1, DD_ - 1);

    float part = qv * trisample(kb, z0, z1, y0, y1, x0, x1, wz, wy, wx);
    for (int off = 16; off; off >>= 1) part += __shfl_xor(part, off, 32);
    float sc = part + bias[o];

    float mn = fmaxf(m, sc);
    float corr = __expf(m - mn);
    float p = __expf(sc - mn);
    float vs = trisample(vb, z0, z1, y0, y1, x0, x1, wz, wy, wx);
    l = l * corr + p;
    acc = acc * corr + p * vs;
    m = mn;
  }
  attn_t[(((size_t)b * NH_ + head) * S_ + s) * HD_ + lane] = acc / l;
}

// ======= final 1x1 conv GEMM, fusing (attn transpose + lepe) B-load =========
__global__ __launch_bounds__(256) void out_gemm_kernel(
    const float* __restrict__ attn_t, const float* __restrict__ lepe,
    const float* __restrict__ pw, const float* __restrict__ pbias,
    float* __restrict__ out)
{
  const int lane = threadIdx.x & 31;
  const int gw   = blockIdx.x * 8 + (threadIdx.x >> 5);
  if (gw >= B_ * 12 * 1024) return;
  const int nt = gw & 1023;
  const int mt = (gw >> 10) % 12;
  const int b  = gw / (12 * 1024);
  const int n     = lane & 15;
  const int khalf = (lane >> 4) << 1;
  const int sbase = nt << 4;
  const int mbase = mt << 4;
  const int ss = sbase + n;

  const float* pwr = pw + (size_t)(mbase + n) * C_ + khalf;
  v8f acc = {};
  for (int k0 = 0; k0 < C_; k0 += 4) {
    int c0 = k0 + khalf, c1 = c0 + 1;
    v2f bx;
    bx.x = attn_t[(((size_t)b * NH_ + (c0 >> 5)) * S_ + ss) * HD_ + (c0 & 31)]
         + lepe[((size_t)b * C_ + c0) * S_ + ss];
    bx.y = attn_t[(((size_t)b * NH_ + (c1 >> 5)) * S_ + ss) * HD_ + (c1 & 31)]
         + lepe[((size_t)b * C_ + c1) * S_ + ss];
    v2f a = *(const v2f*)(pwr + k0);
    acc = __builtin_amdgcn_wmma_f32_16x16x4_f32(false, a, false, bx, (short)0, acc, false, false);
  }
  const int mhi = (lane >> 4) << 3;
  for (int r = 0; r < 8; ++r) {
    int mm = mbase + r + mhi;
    out[((size_t)b * C_ + mm) * S_ + ss] = acc[r] + pbias[mm];
  }
}

// =============== deterministic reductions: offset_mag / energy ==============
__global__ __launch_bounds__(256) void offmag_kernel(
    const float* __restrict__ offs_o, float* __restrict__ mag_o, float* __restrict__ guide_o)
{
  int bh = blockIdx.x;                          // 0..11 -> (b, head)
  const float* op = offs_o + (size_t)bh * S_ * 3;
  float sum = 0.f;
  for (int s = threadIdx.x; s < S_; s += 256) {
    float x = op[s * 3], y = op[s * 3 + 1], z = op[s * 3 + 2];
    sum += sqrtf(x * x + y * y + z * z);
  }
  __shared__ float red[256];
  red[threadIdx.x] = sum; __syncthreads();
  for (int st = 128; st; st >>= 1) {
    if (threadIdx.x < st) red[threadIdx.x] += red[threadIdx.x + st];
    __syncthreads();
  }
  if (threadIdx.x == 0) {
    float mres = red[0] * (1.f / S_);
    int b = bh / NH_, h = bh % NH_;
    mag_o[bh] = mres;
    guide_o[b * 8 + h] = mres;
  }
}

__global__ __launch_bounds__(256) void energy_kernel(
    const float* __restrict__ attn_t, float* __restrict__ be_o, float* __restrict__ guide_o)
{
  int bb = blockIdx.x;                          // 0..3 -> (b, branch)
  int b = bb >> 1, br = bb & 1;
  const float* p = attn_t + (((size_t)b * NH_ + br * 3) * S_) * HD_;
  const size_t n = (size_t)3 * S_ * HD_;
  float sum = 0.f;
  for (size_t i = threadIdx.x; i < n; i += 256) sum += fabsf(p[i]);
  __shared__ float red[256];
  red[threadIdx.x] = sum; __syncthreads();
  for (int st = 128; st; st >>= 1) {
    if (threadIdx.x < st) red[threadIdx.x] += red[threadIdx.x + st];
    __syncthreads();
  }
  if (threadIdx.x == 0) {
    float e = red[0] * (1.f / (float)n);
    be_o[b * 2 + br] = e;
    guide_o[b * 8 + 6 + br] = e;
  }
}

// ================================ launcher ==================================
extern "C" void kernel_launch(void* const* d_in, const int* in_sizes, int n_in,
                              void* d_out, int out_size, void* d_ws, size_t ws_size,
                              hipStream_t stream) {
  (void)in_sizes; (void)n_in; (void)out_size; (void)ws_size;
  const float* x     = (const float*)d_in[0];
  const float* qw    = (const float*)d_in[1];
  const float* qb    = (const float*)d_in[2];
  const float* kw    = (const float*)d_in[3];
  const float* kb    = (const float*)d_in[4];
  const float* vw    = (const float*)d_in[5];
  const float* vb    = (const float*)d_in[6];
  const float* pw    = (const float*)d_in[7];
  const float* pb    = (const float*)d_in[8];
  const float* odw   = (const float*)d_in[9];
  const float* odb   = (const float*)d_in[10];
  const float* lnw   = (const float*)d_in[11];
  const float* lnb   = (const float*)d_in[12];
  const float* opw   = (const float*)d_in[13];
  const float* opb   = (const float*)d_in[14];
  const float* rpew  = (const float*)d_in[15];
  const float* rpeb  = (const float*)d_in[16];
  const float* bias0 = (const float*)d_in[17];
  const float* bias1 = (const float*)d_in[18];

  float* ws     = (float*)d_ws;                 // needs 6 planes = ~151 MB
  float* q_std  = ws;
  float* k_t    = ws + 1 * PLANE_;
  float* v_t    = ws + 2 * PLANE_;
  float* lepe   = ws + 3 * PLANE_;
  float* t      = ws + 4 * PLANE_;
  float* attn_t = ws + 5 * PLANE_;

  float* out     = (float*)d_out;               // [2,192,16,32,32]
  float* offs_o  = out + 6291456;               // [2,6,16,32,32,3]
  float* mag_o   = out + 6881280;               // [2,6]
  float* be_o    = out + 6881292;               // [2,2]
  float* guide_o = out + 6881296;               // [2,8]

  qkv_gemm_kernel<<<3072, 256, 0, stream>>>(x, qw, qb, kw, kb, vw, vb, q_std, k_t, v_t);
  dwconv_kernel<<<(B_ * C_ * S_ + 255) / 256, 256, 0, stream>>>(q_std, odw, odb, rpew, rpeb, t, lepe);
  ln_gelu_kernel<<<4096, 256, 0, stream>>>(t, lnw, lnb);
  offs_kernel<<<(18 * B_ * S_ + 255) / 256, 256, 0, stream>>>(t, opw, opb, offs_o);
  attn_kernel<<<24576, 256, 0, stream>>>(q_std, k_t, v_t, offs_o, bias0, bias1, attn_t);
  out_gemm_kernel<<<3072, 256, 0, stream>>>(attn_t, lepe, pw, pb, out);
  offmag_kernel<<<12, 256, 0, stream>>>(offs_o, mag_o, guide_o);
  energy_kernel<<<4, 256, 0, stream>>>(attn_t, be_o, guide_o);
}